// ModelNew_3556232922267
// MI455X (gfx1250) — compile-verified
//
#include <hip/hip_runtime.h>

// CDNA5 / gfx1250. wave32. fp32 WMMA path: v_wmma_f32_16x16x4_f32.

typedef __attribute__((ext_vector_type(2))) float v2f;
typedef __attribute__((ext_vector_type(8))) float v8f;

#define CI    8
#define CO    8
#define KS    3
#define SS    64          // input spatial
#define OS    66          // output spatial (full conv: S + K - 1)
#define KTOT  216         // CI * 27
#define WPAD  68          // staged input row length: w_in in [-2,65] -> [0,67]
#define NWAVE 5           // 5 waves x 16 outputs covers 66 (+ masked tail)
#define NTHR  (NWAVE * 32)

// LDS offset of A element for reduction index k = ((ci*3+kd)*3+kh)*3+kw.
// Constant-folds when called with unroll-constant k.
__device__ __forceinline__ constexpr int a_offs(int k) {
    int ci = k / 27, r2 = k % 27;
    int kd = r2 / 9, kh = (r2 / 3) % 3, kw = r2 % 3;
    return ((ci * 3 + kd) * 3 + kh) * WPAD + kw;
}

__global__ __launch_bounds__(NTHR)
void conv3d_bn_relu_mean_wmma(const float* __restrict__ x,
                              const float* __restrict__ weight,
                              const float* __restrict__ bias,
                              const float* __restrict__ rmean,
                              const float* __restrict__ rvar,
                              const float* __restrict__ gamma,
                              const float* __restrict__ beta,
                              float* __restrict__ ws) {
    __shared__ float sB[KTOT * 16];               // flipped weights, zero-padded to 16 cols
    __shared__ float sX[CI * 3 * 3 * WPAD + 16];  // +16 pad: masked tail lanes may overread
    __shared__ float sAcc[NWAVE][CO];

    const int tid  = threadIdx.x;
    const int lane = tid & 31;
    const int wid  = tid >> 5;
    const int n    = blockIdx.x / OS;
    const int d    = blockIdx.x % OS;

    // ---- stage flipped weights as B[k][col] (cols 8..15 = 0) ----
    for (int i = tid; i < KTOT * 16; i += NTHR) {
        int k = i >> 4, col = i & 15;
        float v = 0.f;
        if (col < CO) {
            int ci = k / 27, r2 = k % 27;
            int kd = r2 / 9, kh = (r2 / 3) % 3, kw = r2 % 3;
            v = weight[(((col * CI + ci) * KS + (2 - kd)) * KS + (2 - kh)) * KS + (2 - kw)];
        }
        sB[i] = v;
    }
    if (tid < 16) sX[CI * 3 * 3 * WPAD + tid] = 0.f;   // zero the overread pad

    const int r    = lane & 15;   // A: M row index; B/D: N column index
    const int half = lane >> 4;   // K-half selector per 32-bit WMMA layout
    const int w0   = wid * 16;    // output-w tile base for this wave
    float partial  = 0.f;         // running ReLU-sum for column r (valid when r < CO)

    // per-column BN constants (inference)
    float c_bias = 0.f, c_g = 0.f, c_mean = 0.f, c_beta = 0.f;
    if (r < CO) {
        c_bias = bias[r];
        c_g    = rsqrtf(rvar[r] + 1e-5f) * gamma[r];
        c_mean = rmean[r];
        c_beta = beta[r];
    }

    const size_t xbase = (size_t)n * CI * SS * SS * SS;

    for (int h = 0; h < OS; ++h) {
        __syncthreads();  // protect previous iteration's sX readers
        // ---- stage x rows (zi,hi in [d-2..d]x[h-2..h], zero-filled halo) ----
        for (int i = tid; i < CI * 3 * 3 * WPAD; i += NTHR) {
            int wp = i % WPAD;
            int t2 = i / WPAD;
            int kh = t2 % 3;
            int kd = (t2 / 3) % 3;
            int ci = t2 / 9;
            int zi = d - 2 + kd;
            int hi = h - 2 + kh;
            int wi = wp - 2;
            float v = 0.f;
            if ((unsigned)zi < SS && (unsigned)hi < SS && (unsigned)wi < SS) {
                const float* p = x + xbase + (((size_t)ci * SS + zi) * SS + hi) * SS + wi;
                v = *p;
                if (kh == 2 && hi + 1 < SS)            // pull next h's new row into cache
                    __builtin_prefetch(p + SS, 0, 3);  // -> global_prefetch_b8
            }
            sX[i] = v;
        }
        __syncthreads();

        // ---- implicit GEMM: D(16x16) += A(16x4) x B(4x16), K = 216 ----
        v8f acc = {0.f, 0.f, 0.f, 0.f, 0.f, 0.f, 0.f, 0.f};
        #pragma unroll
        for (int t = 0; t < KTOT / 4; ++t) {
            // lane-half picks K rows {4t,4t+1} or {4t+2,4t+3}: select of constants
            int oA0 = half ? a_offs(4 * t + 2) : a_offs(4 * t + 0);
            int oA1 = half ? a_offs(4 * t + 3) : a_offs(4 * t + 1);
            v2f a, b;
            a.x = sX[oA0 + w0 + r];
            a.y = sX[oA1 + w0 + r];
            int bb = (4 * t + 2 * half) * 16 + r;
            b.x = sB[bb];
            b.y = sB[bb + 16];
            acc = __builtin_amdgcn_wmma_f32_16x16x4_f32(
                      false, a, false, b, (short)0, acc, false, false);
        }

        // ---- bias*SCALE, BN, ReLU, accumulate spatial sum ----
        if (r < CO) {
            #pragma unroll
            for (int v = 0; v < 8; ++v) {
                int m = v + 8 * half;          // D row in this lane's VGPR v
                if (w0 + m < OS) {             // mask tail of wave 4
                    float y = (acc[v] + c_bias) * 2.0f;
                    y = (y - c_mean) * c_g + c_beta;
                    partial += fmaxf(y, 0.f);
                }
            }
        }
    }

    // combine half-lanes (lane co <-> lane co+16) via ds_swizzle XOR-16
    int pi = __float_as_int(partial);
    int qi = __builtin_amdgcn_ds_swizzle(pi, 0x401f);  // xor_mask=0x10, and_mask=0x1f
    partial += __int_as_float(qi);

    if (lane < CO) sAcc[wid][lane] = partial;
    __syncthreads();
    if (tid < CO) {
        float s = 0.f;
        #pragma unroll
        for (int wv = 0; wv < NWAVE; ++wv) s += sAcc[wv][tid];  // fixed order: deterministic
        ws[(size_t)blockIdx.x * CO + tid] = s;
    }
}

// Reduce 66 per-(n,d) partials per (n,co) and divide by 66^3.
__global__ __launch_bounds__(128)
void finalize_mean(const float* __restrict__ ws, float* __restrict__ out) {
    int t = threadIdx.x;
    if (t < 16 * CO) {
        int n = t >> 3, co = t & 7;
        float s = 0.f;
        for (int dd = 0; dd < OS; ++dd)
            s += ws[(size_t)(n * OS + dd) * CO + co];
        out[t] = s * (1.0f / (float)(OS * OS * OS));
    }
}

extern "C" void kernel_launch(void* const* d_in, const int* in_sizes, int n_in,
                              void* d_out, int out_size, void* d_ws, size_t ws_size,
                              hipStream_t stream) {
    const float* x     = (const float*)d_in[0];
    const float* w     = (const float*)d_in[1];
    const float* bias  = (const float*)d_in[2];
    const float* rmean = (const float*)d_in[3];
    const float* rvar  = (const float*)d_in[4];
    const float* gamma = (const float*)d_in[5];
    const float* beta  = (const float*)d_in[6];
    float* ws  = (float*)d_ws;     // uses 16*66*8*4 = 33,792 bytes
    float* out = (float*)d_out;    // 128 floats

    conv3d_bn_relu_mean_wmma<<<dim3(16 * OS), dim3(NTHR), 0, stream>>>(
        x, w, bias, rmean, rvar, gamma, beta, ws);
    finalize_mean<<<dim3(1), dim3(128), 0, stream>>>(ws, out);
}